// MM_GCN_75084618268709
// MI455X (gfx1250) — compile-verified
//
#include <hip/hip_runtime.h>
#include <math.h>

// ---- static problem dims (dia_len is statically all-64 in the reference) ----
#define NDIA 48
#define NU   3072          // utterances
#define N3   9216          // 3*NU nodes
#define DD   512           // feature dim == hidden dim
#define SCALE_F 0.99999f
#define INV_PI 0.31830988618379067f

typedef __attribute__((ext_vector_type(2))) float v2f;
typedef __attribute__((ext_vector_type(8))) float v8f;

// D = A(16x4) * B(4x16) + C, fp32 WMMA (V_WMMA_F32_16X16X4_F32)
__device__ __forceinline__ v8f wmma4(v2f a, v2f b, v8f c) {
  return __builtin_amdgcn_wmma_f32_16x16x4_f32(false, a, false, b, (short)0, c,
                                               false, false);
}

// ---------------------------------------------------------------------------
// K1: X = [a ; v ; l + W_spk[argmax(qmask)]]   (row-major [9216,512])
__global__ void k_init_x(const float* __restrict__ a, const float* __restrict__ v,
                         const float* __restrict__ l, const float* __restrict__ qmask,
                         const float* __restrict__ W_spk, float* __restrict__ X) {
  int idx = blockIdx.x * 256 + threadIdx.x;        // < 9216*512
  int row = idx >> 9;
  int col = idx & 511;
  float val;
  if (row < NU) {
    val = a[idx];
  } else if (row < 2 * NU) {
    val = v[(row - NU) * DD + col];
  } else {
    int u = row - 2 * NU;
    int dia = u >> 6, pos = u & 63;
    int qb = (pos * NDIA + dia) * 2;
    int spk = (qmask[qb + 1] > qmask[qb]) ? 1 : 0;     // argmax, first-on-tie
    val = l[u * DD + col] + W_spk[spk * DD + col];
  }
  X[idx] = val;
}

// K2: per-row inverse L2 norm (one block per row)
__global__ void k_invnorm(const float* __restrict__ X, float* __restrict__ inv) {
  __shared__ float red[256];
  const float* xr = X + (size_t)blockIdx.x * DD;
  float s = 0.f;
  for (int c = threadIdx.x; c < DD; c += 256) { float t = xr[c]; s += t * t; }
  red[threadIdx.x] = s; __syncthreads();
  for (int off = 128; off > 0; off >>= 1) {
    if (threadIdx.x < off) red[threadIdx.x] += red[threadIdx.x + off];
    __syncthreads();
  }
  if (threadIdx.x == 0) inv[blockIdx.x] = rsqrtf(red[0]);
}

// K3: within-modality per-dialogue 64x64 angular similarity via fp32 WMMA.
// One block per (modality,dialogue); 8 waves, 2 16x16 tiles each; K=512.
__global__ void k_sim(const float* __restrict__ X, const float* __restrict__ inv,
                      float* __restrict__ S) {
  int md = blockIdx.x;                      // 0..143
  int m = md / NDIA, d = md % NDIA;
  const float* Xb   = X   + ((size_t)m * NU + d * 64) * DD;
  const float* invb = inv + m * NU + d * 64;
  int wave = threadIdx.x >> 5, lane = threadIdx.x & 31;
  int lm = lane & 15, lk2 = (lane >> 4) * 2;
  int tr  = wave & 3;                       // row tile 0..3
  int tc0 = (wave >> 2) * 2;                // col tile base 0 or 2
  v8f acc0 = {}; v8f acc1 = {};
  int arow = tr * 16 + lm;
  int n0 = tc0 * 16 + lm, n1 = n0 + 16;
  for (int k0 = 0; k0 < DD; k0 += 4) {
    int kb = k0 + lk2;
    v2f af; af.x = Xb[arow * DD + kb]; af.y = Xb[arow * DD + kb + 1];
    v2f b0; b0.x = Xb[n0 * DD + kb];   b0.y = Xb[n0 * DD + kb + 1];  // B = Xb^T
    v2f b1; b1.x = Xb[n1 * DD + kb];   b1.y = Xb[n1 * DD + kb + 1];
    acc0 = wmma4(af, b0, acc0);
    acc1 = wmma4(af, b1, acc1);
  }
  float* Sb = S + (size_t)md * 64 * 64;
  for (int t = 0; t < 2; ++t) {
    v8f acc = t ? acc1 : acc0;
    int gn = (tc0 + t) * 16 + lm;
    float invn = invb[gn];
    for (int r = 0; r < 8; ++r) {
      int gm = tr * 16 + (lane >> 4) * 8 + r;
      float cs = acc[r] * invb[gm] * invn * SCALE_F;
      cs = fminf(fmaxf(cs, -SCALE_F), SCALE_F);
      Sb[gm * 64 + gn] = 1.0f - acosf(cs) * INV_PI;
    }
  }
}

// K4: cross-modality same-utterance angular similarity (3 unordered pairs)
__global__ void k_cross(const float* __restrict__ X, const float* __restrict__ inv,
                        float* __restrict__ CX) {
  __shared__ float red[128];
  int pid = blockIdx.x;                 // pair*NU + u, pair in {0,1,2}
  int pair = pid / NU, u = pid % NU;
  int m = (pair == 2) ? 1 : 0;
  int n = (pair == 0) ? 1 : 2;
  const float* xm = X + ((size_t)m * NU + u) * DD;
  const float* xn = X + ((size_t)n * NU + u) * DD;
  float s = 0.f;
  for (int c = threadIdx.x; c < DD; c += 128) s += xm[c] * xn[c];
  red[threadIdx.x] = s; __syncthreads();
  for (int off = 64; off > 0; off >>= 1) {
    if (threadIdx.x < off) red[threadIdx.x] += red[threadIdx.x + off];
    __syncthreads();
  }
  if (threadIdx.x == 0) {
    float cs = red[0] * inv[m * NU + u] * inv[n * NU + u] * SCALE_F;
    cs = fminf(fmaxf(cs, -SCALE_F), SCALE_F);
    float val = 1.0f - acosf(cs) * INV_PI;
    CX[(m * 3 + n) * NU + u] = val;
    CX[(n * 3 + m) * NU + u] = val;
  }
}

// K5: degree -> dinv = rsqrt(row sum of block + 2 cross entries)
__global__ void k_degree(const float* __restrict__ S, const float* __restrict__ CX,
                         float* __restrict__ dinv) {
  int i = blockIdx.x * 256 + threadIdx.x;      // node id < 9216
  int m = i / NU, u = i % NU, d = u >> 6, pos = u & 63;
  const float* row = S + ((size_t)(m * NDIA + d) * 64 + pos) * 64;
  float s = 0.f;
  for (int j = 0; j < 64; ++j) s += row[j];
  for (int n = 0; n < 3; ++n) if (n != m) s += CX[(m * 3 + n) * NU + u];
  dinv[i] = rsqrtf(s);
}

// K6: symmetric normalization of S blocks and CX diagonals in place
__global__ void k_norm(float* __restrict__ S, float* __restrict__ CX,
                       const float* __restrict__ dinv) {
  int e = blockIdx.x * 256 + threadIdx.x;      // < 589824 + 27648
  if (e < 589824) {
    int md = e >> 12, ij = e & 4095;
    int i = ij >> 6, j = ij & 63;
    int m = md / NDIA, d = md % NDIA;
    S[e] *= dinv[m * NU + d * 64 + i] * dinv[m * NU + d * 64 + j];
  } else {
    int e2 = e - 589824;                        // < 9*3072
    int mn = e2 / NU, u = e2 % NU;
    int m = mn / 3, n = mn % 3;
    if (m != n) CX[e2] *= dinv[m * NU + u] * dinv[n * NU + u];
  }
}

// K7: C[9216,512] = A[9216,512] @ B[512,512] with epilogue.
// mode 0: C = relu(G + bias[col])                        (h0)
// mode 1: C = relu(theta*G + (1-theta)*SUP[row,col])     (GCNII layer)
// grid = (8, 72); block = 256 (8 waves); wave tile = 16 rows x 64 cols.
__global__ void k_gemm512(const float* __restrict__ A, const float* __restrict__ B,
                          float* __restrict__ C, const float* __restrict__ bias,
                          const float* __restrict__ SUP, float theta, int mode) {
  int wave = threadIdx.x >> 5, lane = threadIdx.x & 31;
  int lm = lane & 15, lk2 = (lane >> 4) * 2;
  int rowBase = blockIdx.y * 128 + wave * 16;
  int colBase = blockIdx.x * 64;
  v8f acc[4] = {};
  const float* Arow = A + (size_t)(rowBase + lm) * DD;
  for (int k0 = 0; k0 < DD; k0 += 4) {
    int kb = k0 + lk2;
    v2f af; af.x = Arow[kb]; af.y = Arow[kb + 1];
#pragma unroll
    for (int t = 0; t < 4; ++t) {
      int n = colBase + t * 16 + lm;
      v2f bf; bf.x = B[(size_t)kb * DD + n]; bf.y = B[(size_t)(kb + 1) * DD + n];
      acc[t] = wmma4(af, bf, acc[t]);
    }
  }
#pragma unroll
  for (int t = 0; t < 4; ++t) {
    int gn = colBase + t * 16 + lm;
#pragma unroll
    for (int r = 0; r < 8; ++r) {
      int gm = rowBase + (lane >> 4) * 8 + r;
      float g = acc[t][r], o;
      if (mode == 0) {
        o = fmaxf(g + bias[gn], 0.0f);
      } else {
        float s = SUP[(size_t)gm * DD + gn];
        o = fmaxf(theta * g + (1.0f - theta) * s, 0.0f);
      }
      C[(size_t)gm * DD + gn] = o;
    }
  }
}

// K8: block-sparse adjacency apply fused with GCNII support combine:
// SUP[m,u,:] = 0.9*( S'_{m,d} @ h_block + sum_{n!=m} c'[m,n,u]*h[n,u,:] ) + 0.1*h0
// grid = 288 (= 144 (m,d) pairs * 2 column halves); 8 waves per block;
// wave tile = 16 rows x 128 cols; K = 64.
__global__ void k_adj(const float* __restrict__ S, const float* __restrict__ CX,
                      const float* __restrict__ h, const float* __restrict__ h0,
                      float* __restrict__ SUP) {
  int bid = blockIdx.x;
  int pairIdx = bid >> 1, half = bid & 1;
  int m = pairIdx / NDIA, d = pairIdx % NDIA;
  int wave = threadIdx.x >> 5, lane = threadIdx.x & 31;
  int lm = lane & 15, lk2 = (lane >> 4) * 2;
  int mtile = wave & 3;
  int nbase = half * 256 + (wave >> 2) * 128;
  const float* Sb = S + (size_t)(m * NDIA + d) * 4096;
  const float* hb = h + ((size_t)m * NU + d * 64) * DD;
  v8f acc[8] = {};
  int arow = mtile * 16 + lm;
  for (int k0 = 0; k0 < 64; k0 += 4) {
    int kb = k0 + lk2;
    v2f af; af.x = Sb[arow * 64 + kb]; af.y = Sb[arow * 64 + kb + 1];
#pragma unroll
    for (int t = 0; t < 8; ++t) {
      int n = nbase + t * 16 + lm;
      v2f bf; bf.x = hb[(size_t)kb * DD + n]; bf.y = hb[(size_t)(kb + 1) * DD + n];
      acc[t] = wmma4(af, bf, acc[t]);
    }
  }
  int o1 = (m == 0) ? 1 : 0;
  int o2 = (m == 2) ? 1 : 2;
#pragma unroll
  for (int t = 0; t < 8; ++t) {
    int gn = nbase + t * 16 + lm;
#pragma unroll
    for (int r = 0; r < 8; ++r) {
      int u = d * 64 + mtile * 16 + (lane >> 4) * 8 + r;
      size_t rowM = (size_t)(m * NU + u) * DD;
      float y = acc[t][r]
              + CX[(m * 3 + o1) * NU + u] * h[(size_t)(o1 * NU + u) * DD + gn]
              + CX[(m * 3 + o2) * NU + u] * h[(size_t)(o2 * NU + u) * DD + gn];
      SUP[rowM + gn] = 0.9f * y + 0.1f * h0[rowM + gn];
    }
  }
}

// K9: out[u, m*1024 + {feat 0..511 = X, 512..1023 = h}]
__global__ void k_out(const float* __restrict__ X, const float* __restrict__ h,
                      float* __restrict__ out) {
  size_t idx = (size_t)blockIdx.x * 256 + threadIdx.x;   // < 3072*3072
  int u = (int)(idx / 3072);
  int rem = (int)(idx % 3072);
  int m = rem >> 10, q = rem & 1023;
  float val = (q < 512) ? X[(size_t)(m * NU + u) * DD + q]
                        : h[(size_t)(m * NU + u) * DD + (q - 512)];
  out[idx] = val;
}

// ---------------------------------------------------------------------------
extern "C" void kernel_launch(void* const* d_in, const int* in_sizes, int n_in,
                              void* d_out, int out_size, void* d_ws, size_t ws_size,
                              hipStream_t stream) {
  const float* a     = (const float*)d_in[0];
  const float* v     = (const float*)d_in[1];
  const float* l     = (const float*)d_in[2];
  const float* qmask = (const float*)d_in[3];
  // d_in[4] = dia_len (int64) — statically all 64, unused
  const float* W_spk = (const float*)d_in[5];
  const float* W0    = (const float*)d_in[6];
  const float* b0    = (const float*)d_in[7];
  const float* Wg    = (const float*)d_in[8];
  float* out = (float*)d_out;

  // workspace layout (floats): ~78.1 MB total
  float* ws   = (float*)d_ws;
  float* X    = ws;                    // 9216*512
  float* H0   = X   + 4718592;         // 9216*512
  float* H    = H0  + 4718592;         // 9216*512
  float* SUP  = H   + 4718592;         // 9216*512
  float* S    = SUP + 4718592;         // 3*48*64*64
  float* CX   = S   + 589824;          // 9*3072
  float* INV  = CX  + 27648;           // 9216
  float* DINV = INV + 9216;            // 9216
  (void)in_sizes; (void)n_in; (void)out_size; (void)ws_size;

  k_init_x <<<18432, 256, 0, stream>>>(a, v, l, qmask, W_spk, X);
  k_invnorm<<<N3,    256, 0, stream>>>(X, INV);
  k_sim    <<<144,   256, 0, stream>>>(X, INV, S);
  k_cross  <<<3*NU,  128, 0, stream>>>(X, INV, CX);
  k_degree <<<36,    256, 0, stream>>>(S, CX, DINV);
  k_norm   <<<2412,  256, 0, stream>>>(S, CX, DINV);          // 589824+27648 = 617472

  k_gemm512<<<dim3(8, 72), 256, 0, stream>>>(X, W0, H0, b0, nullptr, 0.f, 0);

  const float thetas[2] = {0.4054651081081644f, 0.22314355131420976f}; // log(1.5), log(1.25)
  const float* hin = H0;
  for (int layer = 0; layer < 2; ++layer) {
    k_adj    <<<288, 256, 0, stream>>>(S, CX, hin, H0, SUP);
    k_gemm512<<<dim3(8, 72), 256, 0, stream>>>(SUP, Wg + (size_t)layer * DD * DD, H,
                                               nullptr, SUP, thetas[layer], 1);
    hin = H;
  }
  k_out<<<36864, 256, 0, stream>>>(X, H, out);
}